// BasicBlock7_4947802325430
// MI455X (gfx1250) — compile-verified
//
#include <hip/hip_runtime.h>
#include <math.h>

typedef __attribute__((ext_vector_type(2))) float v2f;
typedef __attribute__((ext_vector_type(8))) float v8f;

#define B_    4
#define CIN_  128
#define COUT_ 128
#define H_    56
#define W_    56
#define HW_   (H_ * W_)          // 3136
#define NTOT  (B_ * HW_)         // 12544 = 392 * 32
#define EPS_  1e-5f
#define LDSTR 34                 // padded row stride (dwords): conflict-free, even

// ---------------------------------------------------------------------------
// Implicit-GEMM 3x3 conv via V_WMMA_F32_16X16X4_F32.
// Block = 256 threads (8 waves). Wave w computes co-tile [16w, +16) x
// n-tile [32*blockIdx.x, +32) (two 16x16 accumulators sharing A fetches).
// K = CIN*9 = 1152 streamed in 32-wide chunks, double-buffered in LDS.
// ---------------------------------------------------------------------------
__global__ __launch_bounds__(256) void conv3x3_wmma(
    const float* __restrict__ x, const float* __restrict__ w,
    float* __restrict__ out, int pad, int dil)
{
    const int KTOT = CIN_ * 9; // 1152
    __shared__ float bt[2][32 * LDSTR]; // [buf][nn][kk]

    const int tid  = threadIdx.x;
    const int wave = tid >> 5;
    const int lane = tid & 31;
    const int n0   = blockIdx.x * 32;
    const int co0  = wave * 16;

    // B-staging coordinates: 2 n-rows x 2 consecutive k per thread
    const int nn0 = tid >> 4;        // 0..15
    const int nn1 = nn0 + 16;        // 16..31
    const int kk0 = (tid & 15) * 2;  // 0,2,..,30

    const int ng0 = n0 + nn0, ng1 = n0 + nn1;
    const int b0 = ng0 / HW_, rem0 = ng0 % HW_, oy0 = rem0 / W_, ox0 = rem0 % W_;
    const int b1 = ng1 / HW_, rem1 = ng1 % HW_, oy1 = rem1 / W_, ox1 = rem1 % W_;

    const int hi = lane >> 4;        // half-wave K-pair select
    const int ml = lane & 15;

    auto loadChunk = [&](int k0, int buf) {
        float v0[2], v1[2];
        #pragma unroll
        for (int e = 0; e < 2; ++e) {
            const int k  = k0 + kk0 + e;
            const int ci = k / 9;
            const int r  = k % 9;
            const int ky = r / 3;
            const int kx = r % 3;
            const int dyq = ky * dil - pad, dxq = kx * dil - pad;
            int iy = oy0 + dyq, ix = ox0 + dxq;
            v0[e] = (iy >= 0 && iy < H_ && ix >= 0 && ix < W_)
                  ? x[((b0 * CIN_ + ci) * H_ + iy) * W_ + ix] : 0.f;
            iy = oy1 + dyq; ix = ox1 + dxq;
            v1[e] = (iy >= 0 && iy < H_ && ix >= 0 && ix < W_)
                  ? x[((b1 * CIN_ + ci) * H_ + iy) * W_ + ix] : 0.f;
        }
        *(v2f*)&bt[buf][nn0 * LDSTR + kk0] = v2f{v0[0], v0[1]};
        *(v2f*)&bt[buf][nn1 * LDSTR + kk0] = v2f{v1[0], v1[1]};
    };

    v8f acc0 = {}, acc1 = {};
    const int nchunks = KTOT / 32;

    loadChunk(0, 0);
    __syncthreads();
    for (int c = 0; c < nchunks; ++c) {
        const int cur = c & 1;
        if (c + 1 < nchunks) loadChunk((c + 1) * 32, cur ^ 1);
        const int k0 = c * 32;
        #pragma unroll
        for (int s = 0; s < 8; ++s) {
            const int kb = s * 4 + hi * 2;
            v2f a   = *(const v2f*)&w[(co0 + ml) * KTOT + k0 + kb];
            v2f bb0 = *(const v2f*)&bt[cur][ml * LDSTR + kb];
            v2f bb1 = *(const v2f*)&bt[cur][(ml + 16) * LDSTR + kb];
            acc0 = __builtin_amdgcn_wmma_f32_16x16x4_f32(
                false, a, false, bb0, (short)0, acc0, false, false);
            acc1 = __builtin_amdgcn_wmma_f32_16x16x4_f32(
                false, a, false, bb1, (short)0, acc1, false, false);
        }
        __syncthreads();
    }

    // store: VGPR v -> M = v + 8*hi, N = ml (acc0) / ml+16 (acc1)
    #pragma unroll
    for (int half = 0; half < 2; ++half) {
        const int n_out = n0 + half * 16 + ml;
        const int ob    = n_out / HW_;
        const int orem  = n_out % HW_;
        const v8f& acc  = half ? acc1 : acc0;
        #pragma unroll
        for (int v = 0; v < 8; ++v) {
            const int m = v + hi * 8;
            out[(ob * COUT_ + (co0 + m)) * HW_ + orem] = acc[v];
        }
    }
}

// ---------------------------------------------------------------------------
// Deformable 5x5 conv (dil=2, pad=4, stride=1): bilinear corners/weights
// precomputed per (tap, pixel) in LDS, then gather + implicit WMMA GEMM.
// K = CIN*25 = 3200. offs: (B,50,H,W), dy at ch 2t, dx at 2t+1, t = ky*5+kx.
// ---------------------------------------------------------------------------
__global__ __launch_bounds__(256) void deform_wmma(
    const float* __restrict__ x, const float* __restrict__ offs,
    const float* __restrict__ w3, float* __restrict__ out)
{
    const int KTOT = CIN_ * 25; // 3200
    __shared__ float bt[2][32 * LDSTR];
    __shared__ float swgt[25][32][4];
    __shared__ int   scor[25][32][4];

    const int tid  = threadIdx.x;
    const int wave = tid >> 5;
    const int lane = tid & 31;
    const int n0   = blockIdx.x * 32;
    const int co0  = wave * 16;

    const int nn0 = tid >> 4;
    const int nn1 = nn0 + 16;
    const int kk0 = (tid & 15) * 2;

    const int ng0 = n0 + nn0, ng1 = n0 + nn1;
    const int b0 = ng0 / HW_, rem0 = ng0 % HW_, oy0 = rem0 / W_, ox0 = rem0 % W_;
    const int b1 = ng1 / HW_, rem1 = ng1 % HW_, oy1 = rem1 / W_, ox1 = rem1 % W_;

    const int hi = lane >> 4;
    const int ml = lane & 15;

    // ---- precompute bilinear corners & weights per (tap, pixel) ----
    for (int i = tid; i < 25 * 32; i += 256) {
        const int t  = i >> 5;   // tap 0..24
        const int nn = i & 31;
        const int ng = n0 + nn;
        const int bb = ng / HW_;
        const int rm = ng % HW_;
        const int oy = rm / W_, ox = rm % W_;
        const float dy = offs[((bb * 50 + 2 * t) * H_ + oy) * W_ + ox];
        const float dx = offs[((bb * 50 + 2 * t + 1) * H_ + oy) * W_ + ox];
        const float py = dy + (float)((t / 5) * 2) + (float)(oy - 4);
        const float px = dx + (float)((t % 5) * 2) + (float)(ox - 4);
        const float y0f = floorf(py), x0f = floorf(px);
        const float fy = py - y0f, fx = px - x0f;
        const int y0 = (int)y0f, x0 = (int)x0f;
        const int ys[2] = {y0, y0 + 1};
        const int xs[2] = {x0, x0 + 1};
        const float wy[2] = {1.f - fy, fy};
        const float wx[2] = {1.f - fx, fx};
        #pragma unroll
        for (int cy = 0; cy < 2; ++cy)
            #pragma unroll
            for (int cx = 0; cx < 2; ++cx) {
                const int ci2 = cy * 2 + cx;
                const bool ok = ys[cy] >= 0 && ys[cy] < H_ &&
                                xs[cx] >= 0 && xs[cx] < W_;
                scor[t][nn][ci2] = ok ? (ys[cy] * W_ + xs[cx]) : -1;
                swgt[t][nn][ci2] = wy[cy] * wx[cx];
            }
    }
    __syncthreads();

    auto gather = [&](int t, int nn, const float* xp) -> float {
        float s = 0.f;
        #pragma unroll
        for (int c = 0; c < 4; ++c) {
            const int o = scor[t][nn][c];
            if (o >= 0) s += swgt[t][nn][c] * xp[o];
        }
        return s;
    };

    auto loadChunk = [&](int k0, int buf) {
        float v0[2], v1[2];
        #pragma unroll
        for (int e = 0; e < 2; ++e) {
            const int k  = k0 + kk0 + e;
            const int ci = k / 25;
            const int t  = k % 25;
            v0[e] = gather(t, nn0, x + (size_t)(b0 * CIN_ + ci) * HW_);
            v1[e] = gather(t, nn1, x + (size_t)(b1 * CIN_ + ci) * HW_);
        }
        *(v2f*)&bt[buf][nn0 * LDSTR + kk0] = v2f{v0[0], v0[1]};
        *(v2f*)&bt[buf][nn1 * LDSTR + kk0] = v2f{v1[0], v1[1]};
    };

    v8f acc0 = {}, acc1 = {};
    const int nchunks = KTOT / 32;

    loadChunk(0, 0);
    __syncthreads();
    for (int c = 0; c < nchunks; ++c) {
        const int cur = c & 1;
        if (c + 1 < nchunks) loadChunk((c + 1) * 32, cur ^ 1);
        const int k0 = c * 32;
        #pragma unroll
        for (int s = 0; s < 8; ++s) {
            const int kb = s * 4 + hi * 2;
            v2f a   = *(const v2f*)&w3[(co0 + ml) * KTOT + k0 + kb];
            v2f bb0 = *(const v2f*)&bt[cur][ml * LDSTR + kb];
            v2f bb1 = *(const v2f*)&bt[cur][(ml + 16) * LDSTR + kb];
            acc0 = __builtin_amdgcn_wmma_f32_16x16x4_f32(
                false, a, false, bb0, (short)0, acc0, false, false);
            acc1 = __builtin_amdgcn_wmma_f32_16x16x4_f32(
                false, a, false, bb1, (short)0, acc1, false, false);
        }
        __syncthreads();
    }

    #pragma unroll
    for (int half = 0; half < 2; ++half) {
        const int n_out = n0 + half * 16 + ml;
        const int ob    = n_out / HW_;
        const int orem  = n_out % HW_;
        const v8f& acc  = half ? acc1 : acc0;
        #pragma unroll
        for (int v = 0; v < 8; ++v) {
            const int m = v + hi * 8;
            out[(ob * COUT_ + (co0 + m)) * HW_ + orem] = acc[v];
        }
    }
}

// ---------------------------------------------------------------------------
// Small direct conv for the 50-channel offset predictor (3x3, pad=1) + bias.
// ---------------------------------------------------------------------------
__global__ __launch_bounds__(256) void conv_off_k(
    const float* __restrict__ x, const float* __restrict__ w,
    const float* __restrict__ bias, float* __restrict__ out)
{
    const int idx = blockIdx.x * 256 + threadIdx.x;
    const int TOT = B_ * 50 * HW_;
    if (idx >= TOT) return;
    const int ox = idx % W_;
    int t = idx / W_;
    const int oy = t % H_; t /= H_;
    const int oc = t % 50;
    const int b  = t / 50;

    float s = bias[oc];
    for (int ky = 0; ky < 3; ++ky) {
        const int iy = oy + ky - 1;
        if (iy < 0 || iy >= H_) continue;
        for (int kx = 0; kx < 3; ++kx) {
            const int ix = ox + kx - 1;
            if (ix < 0 || ix >= W_) continue;
            const float* xp = x + (b * CIN_) * HW_ + iy * W_ + ix;
            const float* wp = w + (oc * CIN_) * 9 + ky * 3 + kx;
            for (int ci = 0; ci < CIN_; ++ci)
                s += wp[ci * 9] * xp[ci * HW_];
        }
    }
    out[idx] = s;
}

// ---------------------------------------------------------------------------
// Per-channel batch stats: mean and rstd over (B, H, W). grid = COUT_.
// ---------------------------------------------------------------------------
__global__ __launch_bounds__(256) void bn_stats(
    const float* __restrict__ y, float* __restrict__ mean,
    float* __restrict__ rstd)
{
    const int c = blockIdx.x;
    float s = 0.f, s2 = 0.f;
    for (int i = threadIdx.x; i < NTOT; i += 256) {
        const int b = i / HW_;
        const int p = i % HW_;
        const float v = y[(b * COUT_ + c) * HW_ + p];
        s += v; s2 += v * v;
    }
    __shared__ float sh[256], sh2[256];
    sh[threadIdx.x] = s; sh2[threadIdx.x] = s2;
    __syncthreads();
    for (int off = 128; off > 0; off >>= 1) {
        if (threadIdx.x < off) {
            sh[threadIdx.x]  += sh[threadIdx.x + off];
            sh2[threadIdx.x] += sh2[threadIdx.x + off];
        }
        __syncthreads();
    }
    if (threadIdx.x == 0) {
        const float inv = 1.f / (float)NTOT;
        const float m = sh[0] * inv;
        const float var = sh2[0] * inv - m * m;
        mean[c] = m;
        rstd[c] = rsqrtf(var + EPS_);
    }
}

// s = relu(bn(y1)) + relu(bn(y3))
__global__ __launch_bounds__(256) void fuse1(
    const float* __restrict__ y1, const float* __restrict__ y3,
    const float* __restrict__ m1, const float* __restrict__ r1,
    const float* __restrict__ g1, const float* __restrict__ be1,
    const float* __restrict__ m3, const float* __restrict__ r3,
    const float* __restrict__ g3, const float* __restrict__ be3,
    float* __restrict__ s)
{
    const int idx = blockIdx.x * 256 + threadIdx.x;
    const int c = (idx / HW_) % COUT_;
    float a = (y1[idx] - m1[c]) * r1[c] * g1[c] + be1[c];
    float d = (y3[idx] - m3[c]) * r3[c] * g3[c] + be3[c];
    s[idx] = fmaxf(a, 0.f) + fmaxf(d, 0.f);
}

// out = relu(bn(y2) + x)
__global__ __launch_bounds__(256) void fuse2(
    const float* __restrict__ y2, const float* __restrict__ xin,
    const float* __restrict__ m2, const float* __restrict__ r2,
    const float* __restrict__ g2, const float* __restrict__ be2,
    float* __restrict__ out)
{
    const int idx = blockIdx.x * 256 + threadIdx.x;
    const int c = (idx / HW_) % COUT_;
    const float v = (y2[idx] - m2[c]) * r2[c] * g2[c] + be2[c] + xin[idx];
    out[idx] = fmaxf(v, 0.f);
}

extern "C" void kernel_launch(void* const* d_in, const int* in_sizes, int n_in,
                              void* d_out, int out_size, void* d_ws, size_t ws_size,
                              hipStream_t stream) {
    const float* x     = (const float*)d_in[0];
    const float* w1    = (const float*)d_in[1];
    const float* g1    = (const float*)d_in[2];
    const float* b1    = (const float*)d_in[3];
    const float* w_off = (const float*)d_in[4];
    const float* b_off = (const float*)d_in[5];
    const float* w3    = (const float*)d_in[6];
    const float* g3    = (const float*)d_in[7];
    const float* b3    = (const float*)d_in[8];
    const float* w2    = (const float*)d_in[9];
    const float* g2    = (const float*)d_in[10];
    const float* b2    = (const float*)d_in[11];
    float* out = (float*)d_out;

    float* ws   = (float*)d_ws;
    float* off  = ws;                         // 4*50*3136   = 627200
    float* y1   = off + 627200;               // 4*128*3136  = 1605632
    float* y3   = y1 + 1605632;
    float* sbuf = y3 + 1605632;
    float* y2   = sbuf + 1605632;
    float* st   = y2 + 1605632;               // 6*128 stats
    float* m1 = st, *r1 = st + 128, *m3 = st + 256, *r3 = st + 384;
    float* m2 = st + 512, *r2 = st + 640;

    const int NT = B_ * COUT_ * HW_;          // 1605632 = 6272*256

    conv_off_k<<<(B_ * 50 * HW_ + 255) / 256, 256, 0, stream>>>(x, w_off, b_off, off);
    conv3x3_wmma<<<NTOT / 32, 256, 0, stream>>>(x, w1, y1, /*pad=*/2, /*dil=*/2);
    deform_wmma<<<NTOT / 32, 256, 0, stream>>>(x, off, w3, y3);
    bn_stats<<<COUT_, 256, 0, stream>>>(y1, m1, r1);
    bn_stats<<<COUT_, 256, 0, stream>>>(y3, m3, r3);
    fuse1<<<NT / 256, 256, 0, stream>>>(y1, y3, m1, r1, g1, b1, m3, r3, g3, b3, sbuf);
    conv3x3_wmma<<<NTOT / 32, 256, 0, stream>>>(sbuf, w2, y2, /*pad=*/1, /*dil=*/1);
    bn_stats<<<COUT_, 256, 0, stream>>>(y2, m2, r2);
    fuse2<<<NT / 256, 256, 0, stream>>>(y2, x, m2, r2, g2, b2, out);
}